// fusion_gcn_6236292514031
// MI455X (gfx1250) — compile-verified
//
#include <hip/hip_runtime.h>
#include <math.h>

typedef float v2f __attribute__((ext_vector_type(2)));
typedef float v8f __attribute__((ext_vector_type(8)));

#define NNODES 16384
#define NEDGES 163840
#define ETOT   (NEDGES + NNODES)   // with self loops
#define NEG_SLOPE 0.2f
#define TB 256

// ---------------------------------------------------------------------------
// f32 GEMM via V_WMMA_F32_16X16X4_F32.  C = A(MxK) * B(KxN), row-major.
// One wave32 computes a 16x64 strip of C (4 accumulators): A fetched once per
// K-step, 4 independent WMMAs for latency hiding.  Grids divide exactly so
// EXEC is all-1s at every WMMA (ISA requirement).
//   A 16x4 : lanes 0-15 hold M=lane with K{0,1}; lanes 16-31 K{2,3}
//   B 4x16 : lanes 0-15 hold N=lane with K{0,1}; lanes 16-31 K{2,3}
//   C 16x16: VGPR r -> M=r (lanes 0-15) / M=r+8 (lanes 16-31), N=lane&15
// ---------------------------------------------------------------------------
__global__ void gemm_wmma_f32(const float* __restrict__ A,
                              const float* __restrict__ B,
                              float* __restrict__ C,
                              int M, int K, int N)
{
    const int lane = threadIdx.x & 31;
    const int wave = (int)((blockIdx.x * blockDim.x + threadIdx.x) >> 5);
    const int ntn  = N >> 6;                 // 64-wide strips along N
    const int tm   = wave / ntn;
    const int tn   = wave - tm * ntn;
    if (tm * 16 >= M) return;                // wave-uniform guard

    const int half = lane >> 4;              // 0: K{0,1}, 1: K{2,3}
    const int l16  = lane & 15;
    const int koff = half << 1;

    const float* arow = A + (size_t)(tm * 16 + l16) * K + koff;
    const float* brow = B + (size_t)koff * N + tn * 64 + l16;

    v8f acc0 = {}, acc1 = {}, acc2 = {}, acc3 = {};
    #pragma unroll 2
    for (int k = 0; k < K; k += 4) {
        v2f a, b0, b1, b2, b3;
        a.x  = arow[0];      a.y  = arow[1];
        b0.x = brow[0];      b0.y = brow[N];
        b1.x = brow[16];     b1.y = brow[N + 16];
        b2.x = brow[32];     b2.y = brow[N + 32];
        b3.x = brow[48];     b3.y = brow[N + 48];
        acc0 = __builtin_amdgcn_wmma_f32_16x16x4_f32(false, a, false, b0, (short)0, acc0, false, false);
        acc1 = __builtin_amdgcn_wmma_f32_16x16x4_f32(false, a, false, b1, (short)0, acc1, false, false);
        acc2 = __builtin_amdgcn_wmma_f32_16x16x4_f32(false, a, false, b2, (short)0, acc2, false, false);
        acc3 = __builtin_amdgcn_wmma_f32_16x16x4_f32(false, a, false, b3, (short)0, acc3, false, false);
        arow += 4;
        brow += (size_t)4 * N;
    }

    float* crow = C + (size_t)(tm * 16 + half * 8) * N + tn * 64 + l16;
    #pragma unroll
    for (int r = 0; r < 8; ++r) {
        crow[(size_t)r * N +  0] = acc0[r];
        crow[(size_t)r * N + 16] = acc1[r];
        crow[(size_t)r * N + 32] = acc2[r];
        crow[(size_t)r * N + 48] = acc3[r];
    }
}

// ---------------------------------------------------------------------------
// CSR build: degree -> exclusive scan -> scatter -> per-node sort
// ---------------------------------------------------------------------------
__global__ void fill_i32(int* __restrict__ p, int v, int n)
{
    const int i = blockIdx.x * blockDim.x + threadIdx.x;
    if (i < n) p[i] = v;
}

__global__ void degree_kernel(const int* __restrict__ dsts, int* __restrict__ deg, int E_real)
{
    const int e = blockIdx.x * blockDim.x + threadIdx.x;
    if (e >= ETOT) return;
    const int d = (e < E_real) ? dsts[e] : (e - E_real);
    atomicAdd(&deg[d], 1);
}

// single-block Hillis-Steele scan over n (multiple of 256) -> exclusive prefix
__global__ void scan_kernel(const int* __restrict__ deg, int* __restrict__ rowstart, int n)
{
    __shared__ int buf[TB];
    __shared__ int carry;
    if (threadIdx.x == 0) carry = 0;
    __syncthreads();
    for (int base = 0; base < n; base += TB) {
        const int v = deg[base + threadIdx.x];
        int sum = v;
        buf[threadIdx.x] = sum;
        __syncthreads();
        for (int off = 1; off < TB; off <<= 1) {
            const int t = (threadIdx.x >= off) ? buf[threadIdx.x - off] : 0;
            __syncthreads();
            sum += t;
            buf[threadIdx.x] = sum;
            __syncthreads();
        }
        rowstart[base + threadIdx.x] = carry + sum - v;    // exclusive
        __syncthreads();
        if (threadIdx.x == TB - 1) carry += buf[TB - 1];
        __syncthreads();
    }
    if (threadIdx.x == 0) rowstart[n] = carry;
}

__global__ void scatter_kernel(const int* __restrict__ dsts,
                               const int* __restrict__ rowstart,
                               int* __restrict__ cursor,
                               int* __restrict__ eids, int E_real)
{
    const int e = blockIdx.x * blockDim.x + threadIdx.x;
    if (e >= ETOT) return;
    const int d = (e < E_real) ? dsts[e] : (e - E_real);
    const int pos = atomicAdd(&cursor[d], 1);
    eids[rowstart[d] + pos] = e;
}

// insertion sort of each node's segment -> deterministic reduction order
__global__ void sort_kernel(const int* __restrict__ rowstart, int* __restrict__ eids)
{
    const int n = blockIdx.x * blockDim.x + threadIdx.x;
    if (n >= NNODES) return;
    const int beg = rowstart[n], end = rowstart[n + 1];
    for (int i = beg + 1; i < end; ++i) {
        const int key = eids[i];
        int j = i - 1;
        while (j >= beg && eids[j] > key) { eids[j + 1] = eids[j]; --j; }
        eids[j + 1] = key;
    }
}

// ---------------------------------------------------------------------------
// SuperGAT attention + aggregation, one wave per (node, head), single pass
// with online softmax.  xi and att weights held in registers; xj streamed
// (L2-resident).  No float atomics anywhere.
//   alpha_e = leaky_relu((xi.att_i + xj.att_j) * sigmoid(xi.xj))
//   out     = sum_e xj * softmax_dst(alpha)_e  (+ bias)
// ---------------------------------------------------------------------------
template<int CPL>     // channels per lane; C = 32*CPL
__global__ void node_attn_kernel(const float* __restrict__ h,        // [N, H*C]
                                 const int*   __restrict__ srcs,
                                 const int*   __restrict__ rowstart, // [N+1]
                                 const int*   __restrict__ eids,     // [ETOT]
                                 const float* __restrict__ att,      // [H, 2C]
                                 const float* __restrict__ bias,     // [H*C]
                                 float* __restrict__ out,            // [N, H*C]
                                 int E_real, int H)
{
    const int C    = CPL * 32;
    const int wid  = (int)((blockIdx.x * blockDim.x + threadIdx.x) >> 5);
    const int lane = threadIdx.x & 31;
    const int node = wid / H;
    const int hh   = wid - node * H;
    if (node >= NNODES) return;

    const int F = H * C;
    const float* xi_p = h + (size_t)node * F + hh * C;
    const float* wai  = att + (size_t)hh * (2 * C);
    const float* waj  = wai + C;

    float xi[CPL], wj[CPL];
    float ai = 0.0f;
    #pragma unroll
    for (int p = 0; p < CPL; ++p) {
        xi[p] = xi_p[lane + 32 * p];
        wj[p] = waj[lane + 32 * p];
        ai   += xi[p] * wai[lane + 32 * p];
    }
    #pragma unroll
    for (int off = 16; off > 0; off >>= 1) ai += __shfl_xor(ai, off, 32);

    const int beg = rowstart[node];
    const int end = rowstart[node + 1];

    float m = -INFINITY, ssum = 0.0f;
    float acc[CPL];
    #pragma unroll
    for (int p = 0; p < CPL; ++p) acc[p] = 0.0f;

    for (int idx = beg; idx < end; ++idx) {
        const int e = eids[idx];
        const int s = (e < E_real) ? srcs[e] : (e - E_real);
        const float* xj_p = h + (size_t)s * F + hh * C;

        float vj[CPL];
        float dot = 0.0f, aj = 0.0f;
        #pragma unroll
        for (int p = 0; p < CPL; ++p) {
            vj[p] = xj_p[lane + 32 * p];
            dot  += xi[p] * vj[p];
            aj   += wj[p] * vj[p];
        }
        #pragma unroll
        for (int off = 16; off > 0; off >>= 1) {
            dot += __shfl_xor(dot, off, 32);
            aj  += __shfl_xor(aj,  off, 32);
        }
        const float sg = 1.0f / (1.0f + expf(-dot));
        float a = (ai + aj) * sg;
        a = (a >= 0.0f) ? a : NEG_SLOPE * a;

        // online softmax update (wave-uniform scalars)
        const float mn    = fmaxf(m, a);
        const float scale = expf(m - mn);      // first iter: exp(-inf) = 0
        const float ea    = expf(a - mn);
        ssum = ssum * scale + ea;
        #pragma unroll
        for (int p = 0; p < CPL; ++p) acc[p] = acc[p] * scale + vj[p] * ea;
        m = mn;
    }

    const float inv = 1.0f / (ssum + 1e-16f);
    float* o = out + (size_t)node * F + hh * C;
    #pragma unroll
    for (int p = 0; p < CPL; ++p)
        o[lane + 32 * p] = acc[p] * inv + bias[hh * C + lane + 32 * p];
}

// ---------------------------------------------------------------------------
__global__ void gelu_kernel(float* __restrict__ x, int total)
{
    const int i = blockIdx.x * blockDim.x + threadIdx.x;
    if (i >= total) return;
    const float v = x[i];
    x[i] = 0.5f * v * (1.0f + erff(v * 0.70710678118654752f));
}

// ---------------------------------------------------------------------------
extern "C" void kernel_launch(void* const* d_in, const int* in_sizes, int n_in,
                              void* d_out, int out_size, void* d_ws, size_t ws_size,
                              hipStream_t stream)
{
    const float* x    = (const float*)d_in[0];
    const int*   ei   = (const int*)  d_in[1];
    const float* W1   = (const float*)d_in[2];
    const float* att1 = (const float*)d_in[3];
    const float* b1   = (const float*)d_in[4];
    const float* W2   = (const float*)d_in[5];
    const float* att2 = (const float*)d_in[6];
    const float* b2   = (const float*)d_in[7];
    float* out = (float*)d_out;

    const int E_real = in_sizes[1] / 2;      // 163840
    const int* srcs = ei;
    const int* dsts = ei + E_real;

    // workspace carve-up
    float* ws   = (float*)d_ws;
    float* h1   = ws;                              // [N,512]
    float* act  = h1 + (size_t)NNODES * 512;       // [N,512]
    int*   ibuf = (int*)(act + (size_t)NNODES * 512);
    int* deg      = ibuf;                          // [N]
    int* rowstart = deg + NNODES;                  // [N+1]
    int* cursor   = rowstart + NNODES + 1;         // [N]
    int* eids     = cursor + NNODES;               // [ETOT]
    float* h2     = h1;   // reuse: h1 dead after layer-1 attention

    // ---- CSR build (grouped-by-dst edge lists; deterministic via sort) ----
    fill_i32<<<(NNODES + TB - 1) / TB, TB, 0, stream>>>(deg, 0, NNODES);
    fill_i32<<<(NNODES + TB - 1) / TB, TB, 0, stream>>>(cursor, 0, NNODES);
    degree_kernel<<<(ETOT + TB - 1) / TB, TB, 0, stream>>>(dsts, deg, E_real);
    scan_kernel<<<1, TB, 0, stream>>>(deg, rowstart, NNODES);
    scatter_kernel<<<(ETOT + TB - 1) / TB, TB, 0, stream>>>(dsts, rowstart, cursor, eids, E_real);
    sort_kernel<<<(NNODES + TB - 1) / TB, TB, 0, stream>>>(rowstart, eids);

    // ---- layer 1: h1 = x @ W1  (M=16384, K=256, N=512) ----
    {
        const int waves = (NNODES / 16) * (512 / 64);          // 8192
        gemm_wmma_f32<<<waves * 32 / TB, TB, 0, stream>>>(x, W1, h1, NNODES, 256, 512);
    }
    {
        const int waves = NNODES * 4;                          // (node, head)
        node_attn_kernel<4><<<waves * 32 / TB, TB, 0, stream>>>(
            h1, srcs, rowstart, eids, att1, b1, act, E_real, 4);
    }
    gelu_kernel<<<(NNODES * 512 + TB - 1) / TB, TB, 0, stream>>>(act, NNODES * 512);

    // ---- layer 2: h2 = act @ W2  (M=16384, K=512, N=256) ----
    {
        const int waves = (NNODES / 16) * (256 / 64);          // 4096
        gemm_wmma_f32<<<waves * 32 / TB, TB, 0, stream>>>(act, W2, h2, NNODES, 512, 256);
    }
    {
        const int waves = NNODES;                              // H = 1
        node_attn_kernel<8><<<waves * 32 / TB, TB, 0, stream>>>(
            h2, srcs, rowstart, eids, att2, b2, out, E_real, 1);
    }
}